// RotatE_39573828665596
// MI455X (gfx1250) — compile-verified
//
#include <hip/hip_runtime.h>
#include <stdint.h>

#define NUM_E_TOT 100000
#define HID       64
#define BATCHN    32
#define TILE_E    64
#define ROW_F     132           // 128 data floats + 4 pad -> stride%64banks = 4, conflict-free b128
#define GAMMA_F   12.0f
// EMB_RANGE = (12+2)/64 = 0.21875 ; PHASE_SCALE = pi / EMB_RANGE
#define PHASE_SCALE 14.361566416410483f

typedef float v2f __attribute__((ext_vector_type(2)));

// ---------------------------------------------------------------------------
// CDNA5 async global->LDS copy (ASYNCcnt path, GV addressing mode).
// vdst = VGPR holding LDS byte offset, vaddr = 64-bit global address pair.
// ---------------------------------------------------------------------------
static __device__ __forceinline__ void async_copy_b128(uint32_t lds_off, const void* gsrc) {
    asm volatile("global_load_async_to_lds_b128 %0, %1, off"
                 :
                 : "v"(lds_off), "v"((uint64_t)(uintptr_t)gsrc)
                 : "memory");
}

static __device__ __forceinline__ void wait_async_all() {
    asm volatile("s_wait_asynccnt 0x0" ::: "memory");
}

// Direct v_sqrt_f32 (TRANS pipe), no libm fixup code.
static __device__ __forceinline__ float fast_sqrt(float x) {
    return __builtin_amdgcn_sqrtf(x);
}

// ---------------------------------------------------------------------------
// Forced CDNA5 packed-fp32 VOP3P ops (ISA 15.10: opcodes 31/40/41).
// v2f = 64-bit value -> even-aligned VGPR pair, as VOP3P requires.
// ---------------------------------------------------------------------------
static __device__ __forceinline__ v2f pk_sub(v2f a, v2f b) {   // a - b
    v2f d;
    asm("v_pk_add_f32 %0, %1, %2 neg_lo:[0,1] neg_hi:[0,1]"
        : "=v"(d) : "v"(a), "v"(b));
    return d;
}
static __device__ __forceinline__ v2f pk_add(v2f a, v2f b) {
    v2f d;
    asm("v_pk_add_f32 %0, %1, %2" : "=v"(d) : "v"(a), "v"(b));
    return d;
}
static __device__ __forceinline__ v2f pk_mul(v2f a, v2f b) {
    v2f d;
    asm("v_pk_mul_f32 %0, %1, %2" : "=v"(d) : "v"(a), "v"(b));
    return d;
}
static __device__ __forceinline__ v2f pk_fma(v2f a, v2f b, v2f c) {  // a*b + c
    v2f d;
    asm("v_pk_fma_f32 %0, %1, %2, %3" : "=v"(d) : "v"(a), "v"(b), "v"(c));
    return d;
}

// s = (rr-tr)^2 + (ri-ti)^2 per packed lane; sqrt each half.
static __device__ __forceinline__ v2f mag_pair(v2f rr, v2f tr, v2f ri, v2f ti) {
    const v2f d = pk_sub(rr, tr);
    const v2f e = pk_sub(ri, ti);
    const v2f s = pk_fma(e, e, pk_mul(d, d));
    v2f q;
    q.x = fast_sqrt(s.x);
    q.y = fast_sqrt(s.y);
    return q;
}

// ---------------------------------------------------------------------------
// Kernel A: rot_re[32][64] then rot_im[32][64] into d_ws (16 KB).
// 2048 threads total, one (b,k) pair each.
// ---------------------------------------------------------------------------
__global__ void rotate_precompute_kernel(const int* __restrict__ facts,
                                         const float* __restrict__ ent,
                                         const float* __restrict__ rel,
                                         float* __restrict__ rot_ws) {
    const int idx = blockIdx.x * blockDim.x + threadIdx.x;   // 0..2047
    const int b = idx >> 6;
    const int k = idx & 63;
    const int h = facts[b * 3 + 0];
    const int r = facts[b * 3 + 1];
    const float re_h = ent[(size_t)h * (2 * HID) + k];
    const float im_h = ent[(size_t)h * (2 * HID) + HID + k];
    const float ph = rel[(size_t)r * HID + k] * PHASE_SCALE;
    const float cr = cosf(ph);
    const float sr = sinf(ph);
    rot_ws[idx]                 = re_h * cr - im_h * sr;   // rot_re[b][k]
    rot_ws[BATCHN * HID + idx]  = re_h * sr + im_h * cr;   // rot_im[b][k]
}

// ---------------------------------------------------------------------------
// Kernel B: one 64-entity tile per block, all 32 batch rows.
// ---------------------------------------------------------------------------
__global__ void __launch_bounds__(256)
rotate_score_kernel(const float* __restrict__ ent,
                    const float* __restrict__ rot,
                    float* __restrict__ out) {
    __shared__ __align__(16) float s_rot[2 * BATCHN * HID];   // 4096 floats, 16 KB
    __shared__ __align__(16) float s_tail[TILE_E * ROW_F];    // 8448 floats, 33 KB

    const int tid   = threadIdx.x;
    const int tile0 = blockIdx.x * TILE_E;

    // Async-stage rot table: 4096 floats = 1024 x 16B chunks, 4 per thread.
#pragma unroll
    for (int i = 0; i < 4; ++i) {
        const int c = tid + i * 256;
        async_copy_b128((uint32_t)(uintptr_t)&s_rot[c * 4], rot + c * 4);
    }
    // Async-stage entity tile: 64 rows x 128 floats, 32 x 16B chunks per row,
    // scattered into the padded (132-float-stride) LDS layout.
#pragma unroll
    for (int i = 0; i < 8; ++i) {
        const int c = tid + i * 256;        // 0..2047
        const int r = c >> 5;               // entity row in tile
        const int q = c & 31;               // 16B chunk within row
        if (tile0 + r < NUM_E_TOT) {
            async_copy_b128((uint32_t)(uintptr_t)&s_tail[r * ROW_F + q * 4],
                            ent + (size_t)(tile0 + r) * (2 * HID) + q * 4);
        }
    }
    wait_async_all();
    __syncthreads();

    const int e_loc = tid & (TILE_E - 1);   // entity within tile
    const int bq    = tid >> 6;             // which 8-batch-row group

    v2f acc[8];
#pragma unroll
    for (int j = 0; j < 8; ++j) { acc[j].x = 0.0f; acc[j].y = 0.0f; }

    const float* tail_row = &s_tail[e_loc * ROW_F];

#pragma unroll 4
    for (int k4 = 0; k4 < HID / 4; ++k4) {
        const float4 tr4 = *reinterpret_cast<const float4*>(&tail_row[k4 * 4]);
        const float4 ti4 = *reinterpret_cast<const float4*>(&tail_row[HID + k4 * 4]);
        const v2f tr01 = {tr4.x, tr4.y}, tr23 = {tr4.z, tr4.w};
        const v2f ti01 = {ti4.x, ti4.y}, ti23 = {ti4.z, ti4.w};
#pragma unroll
        for (int j = 0; j < 8; ++j) {
            const int b = bq * 8 + j;
            const float4 rr4 = *reinterpret_cast<const float4*>(&s_rot[b * HID + k4 * 4]);
            const float4 ri4 = *reinterpret_cast<const float4*>(&s_rot[BATCHN * HID + b * HID + k4 * 4]);
            const v2f rr01 = {rr4.x, rr4.y}, rr23 = {rr4.z, rr4.w};
            const v2f ri01 = {ri4.x, ri4.y}, ri23 = {ri4.z, ri4.w};
            acc[j] = pk_add(acc[j], mag_pair(rr01, tr01, ri01, ti01));
            acc[j] = pk_add(acc[j], mag_pair(rr23, tr23, ri23, ti23));
        }
    }

    const int e = tile0 + e_loc;
    if (e < NUM_E_TOT) {
#pragma unroll
        for (int j = 0; j < 8; ++j) {
            const float dist = acc[j].x + acc[j].y;
            __builtin_nontemporal_store(GAMMA_F - dist,
                                        &out[(size_t)(bq * 8 + j) * NUM_E_TOT + e]);
        }
    }
}

extern "C" void kernel_launch(void* const* d_in, const int* in_sizes, int n_in,
                              void* d_out, int out_size, void* d_ws, size_t ws_size,
                              hipStream_t stream) {
    const int*   facts = (const int*)d_in[0];    // (32,3) int
    const float* ent   = (const float*)d_in[1];  // (100000, 128) f32
    const float* rel   = (const float*)d_in[2];  // (1000, 64) f32
    float*       out   = (float*)d_out;          // (32, 100000) f32
    float*       rot   = (float*)d_ws;           // 4096 floats scratch

    rotate_precompute_kernel<<<dim3(8), dim3(256), 0, stream>>>(facts, ent, rel, rot);

    const int nblocks = (NUM_E_TOT + TILE_E - 1) / TILE_E;   // 1563
    rotate_score_kernel<<<dim3(nblocks), dim3(256), 0, stream>>>(ent, rot, out);
}